// PeriodicBoundary_18339510354343
// MI455X (gfx1250) — compile-verified
//
#include <hip/hip_runtime.h>

// PeriodicBoundary neighbour list, CDNA5 (gfx1250) version.
// - V_WMMA_F32_16X16X4_F32 computes the 16x16 dist^2 tile directly:
//   A is pre-scaled by -2, C = |p_i|^2 + |q_j|^2, so D = ri2 + rj2 - 2 p.q
// - every per-lane predicate is consumed exactly once, by a ballot; all
//   counting / self-exclusion / summaries are scalar (SGPR) mask arithmetic
// - phase 1 counts rows with s_bcnt on ballot halves; phase 2 replays only
//   non-empty tiles with ballot-rank stream compaction in WMMA D-layout
// - positions staged to LDS with global_load_async_to_lds_b32 (ASYNCcnt path)

typedef __attribute__((ext_vector_type(2))) float v2f;
typedef __attribute__((ext_vector_type(8))) float v8f;

#define CUTOFF2   25.0f
#define SELF_CELL 13      // shift index of (0,0,0) in the 27-cell list
#define MAX_N     1536    // harness-fixed particle count (18 KB in LDS)
#define MAXT      336     // >= ceil(27*(N/16)/8) j-tiles per wave for N=1536

__device__ __forceinline__ unsigned ballot32(bool p) {
#if __has_builtin(__builtin_amdgcn_ballot_w32)
  return __builtin_amdgcn_ballot_w32(p);
#else
  return (unsigned)__ballot(p);
#endif
}

__global__ __launch_bounds__(256)
void pb_fill(int* __restrict__ out, const int* __restrict__ pmaxn,
             int N, int total) {
  int idx = blockIdx.x * blockDim.x + threadIdx.x;
  if (idx >= total) return;
  const int K = *pmaxn;
  const long nk = (long)N * K;
  int v;
  if (idx < nk)            v = -1;  // neighbours pad
  else if (idx < 4 * nk)   v = 1;   // cell_indices pad -> shifts[26] = (1,1,1)
  else                     v = 0;   // actual_max seed for atomicMax
  out[idx] = v;
}

__global__ __launch_bounds__(256)
void pb_neigh(const float* __restrict__ pos, const float* __restrict__ cellm,
              const int* __restrict__ pmaxn, int* __restrict__ out, int N) {
  const int tid      = threadIdx.x;
  const int lane     = tid & 31;
  // wave id is wave-invariant: make it provably uniform -> SALU bookkeeping
  const int w        = __builtin_amdgcn_readfirstlane(tid) >> 5;  // 0..7
  const int halfsel  = lane >> 4;      // 0: lanes 0-15, 1: lanes 16-31
  const int n        = lane & 15;      // WMMA D column of this lane
  const unsigned lanemask = (1u << n) - 1u;
  const int t        = blockIdx.x;     // 16-center row tile
  const float L      = cellm[0];       // diagonal cell length (20.0)
  const int K        = *pmaxn;         // 144
  const int NT       = N >> 4;         // j-tiles per image cell (96)
  const int JT       = 27 * NT;        // total j tiles (2592)

  __shared__ float    s_pos[MAX_N * 3];
  __shared__ int      s_cnt[8][16];
  __shared__ unsigned s_any[8][MAXT];  // per-wave per-tile OR of row ballots

  // Stage positions via the CDNA5 async global->LDS path (GVS addressing).
  for (int i = tid; i < 3 * N && i < 3 * MAX_N; i += blockDim.x) {
    unsigned ldsoff = (unsigned)(size_t)(&s_pos[i]);  // addr[31:0] = LDS offset
    unsigned goff   = (unsigned)(i * 4);
    asm volatile("global_load_async_to_lds_b32 %0, %1, %2"
                 :: "v"(ldsoff), "v"(goff), "s"(pos) : "memory");
  }
  asm volatile("s_wait_asynccnt 0" ::: "memory");
  __syncthreads();

  const int m0 = t << 4;

  // A-matrix (16x4 f32), pre-scaled by -2: VGPR0 = K0|K2, VGPR1 = K1|K3(=0)
  const int mA = m0 + n;
  v2f A;
  A.x = -2.0f * s_pos[3 * mA + (halfsel ? 2 : 0)];
  A.y = halfsel ? 0.0f : -2.0f * s_pos[3 * mA + 1];

  // |p_i|^2 for the 8 D-rows this lane's accumulators cover
  float ri2[8];
#pragma unroll
  for (int v = 0; v < 8; ++v) {
    const int m = m0 + v + (halfsel << 3);
    const float x = s_pos[3 * m], y = s_pos[3 * m + 1], z = s_pos[3 * m + 2];
    ri2[v] = x * x + y * y + z * z;
  }

  // D = A'*B + C with C = ri2 + rj2  ->  D[v] is dist^2 of element (row, col)
  // qoff = 12*(pbase + n): byte offset of this lane's column particle in LDS
  auto tileDist = [&](int qoff, float sxF, float syF, float szF) -> v8f {
    const float* qp = (const float*)((const char*)s_pos + qoff);
    const float qx = qp[0] + sxF;
    const float qy = qp[1] + syF;
    const float qz = qp[2] + szF;
    v2f B;
    B.x = halfsel ? qz : qx;
    B.y = halfsel ? 0.0f : qy;
    const float rj2 = qx * qx + qy * qy + qz * qz;
    v8f C;
#pragma unroll
    for (int v = 0; v < 8; ++v) C[v] = ri2[v] + rj2;
    return __builtin_amdgcn_wmma_f32_16x16x4_f32(
        false, A, false, B, (short)0, C, false, false);
  };

  const int per = (JT + 7) >> 3;  // j tiles per wave (uniform)
  const int jt0 = w * per;
  const int jt1 = (jt0 + per < JT) ? (jt0 + per) : JT;

  const int c0     = jt0 / NT;    // one uniform division at setup
  const int pbase0 = (jt0 - c0 * NT) << 4;

  // ---- Phase 1: scalar (SGPR) row counters from ballot halves ----
  int cntLo[8], cntHi[8];  // uniform: rows v and v+8 of this wave's range
#pragma unroll
  for (int v = 0; v < 8; ++v) { cntLo[v] = 0; cntHi[v] = 0; }
  {
    int c = c0, pbase = pbase0;
    int qoff = 12 * (pbase0 + n);
    int isx = c % 3 - 1, isy = (c / 3) % 3 - 1, isz = c / 9 - 1;
    float sxF = L * (float)isx, syF = L * (float)isy, szF = L * (float)isz;
    for (int jt = jt0; jt < jt1; ++jt) {
      const v8f d2 = tileDist(qoff, sxF, syF, szF);
      // only the diagonal tile of the zero-shift cell can hold self pairs;
      // diagonal elements of accumulator v are ballot bits v and v+24
      const bool diag = (c == SELF_CELL) && (pbase == m0);   // scalar
      const unsigned dk = diag ? 0x01000001u : 0u;           // s_cselect
      unsigned or8 = 0;
#pragma unroll
      for (int v = 0; v < 8; ++v) {
        // i1 consumed exactly once -> stays a bare v_cmp mask
        unsigned bal = ballot32(d2[v] <= CUTOFF2);
        bal &= ~(dk << v);                   // s_lshl + s_andn2
        or8 |= bal;                          // s_or
        cntLo[v] += __popc(bal & 0xFFFFu);   // s_and  + s_bcnt1 + s_add
        cntHi[v] += __popc(bal >> 16);       // s_lshr + s_bcnt1 + s_add
      }
      const int jtl = jt - jt0;
      if (lane == 0 && jtl < MAXT) s_any[w][jtl] = or8;
      pbase += 16;
      qoff  += 192;
      if (pbase >= N) {     // cell advance: shifts updated once per 96 tiles
        pbase = 0; ++c; qoff = 12 * n;
        isx = c % 3 - 1; isy = (c / 3) % 3 - 1; isz = c / 9 - 1;
        sxF = L * (float)isx; syF = L * (float)isy; szF = L * (float)isz;
      }
    }
  }

  // Counts are already per-row scalars: one lane publishes them.
  if (lane == 0) {
#pragma unroll
    for (int v = 0; v < 8; ++v) {
      s_cnt[w][v]     = cntLo[v];
      s_cnt[w][v + 8] = cntHi[v];
    }
  }
  __syncthreads();

  // actual_max = max over centers of unclamped hit count
  if (tid == 0) {
    int mx = 0;
    for (int r = 0; r < 16; ++r) {
      int tot = 0;
      for (int ww = 0; ww < 8; ++ww) tot += s_cnt[ww][r];
      mx = (tot > mx) ? tot : mx;
    }
    atomicMax(out + (size_t)4 * N * K, mx);
  }

  // Exclusive prefix over lower-j waves -> write base for this wave.
  int rcount = 0;  // lanes 0..15 carry the running count for row == lane
  if (lane < 16)
    for (int ww = 0; ww < w; ++ww) rcount += s_cnt[ww][lane];

  // ---- Phase 2: replay only non-empty tiles (scalar skip branch) ----
  {
    int c = c0, pbase = pbase0;
    int qoff = 12 * (pbase0 + n);
    int isx = c % 3 - 1, isy = (c / 3) % 3 - 1, isz = c / 9 - 1;
    float sxF = L * (float)isx, syF = L * (float)isy, szF = L * (float)isz;
    for (int jt = jt0; jt < jt1; ++jt) {
      const int jtl = jt - jt0;
      unsigned any = (jtl < MAXT) ? s_any[w][jtl] : 0xFFFFFFFFu;
      any = (unsigned)__builtin_amdgcn_readfirstlane((int)any);
      if (any) {
        const v8f d2 = tileDist(qoff, sxF, syF, szF);
        const bool diag = (c == SELF_CELL) && (pbase == m0);
        const unsigned dk = diag ? 0x01000001u : 0u;
        const int p = pbase + n;
        unsigned myBal = 0;  // lane v collects row-v ballot (v = 0..7)
#pragma unroll
        for (int v = 0; v < 8; ++v) {
          unsigned bal = ballot32(d2[v] <= CUTOFF2);
          bal &= ~(dk << v);                 // low16: row v, high16: row v+8
          myBal = (lane == v) ? bal : myBal;
          const unsigned half = halfsel ? (bal >> 16) : (bal & 0xFFFFu);
          const int rank = __popc(half & lanemask);
          const int base = __shfl(rcount, v + (halfsel << 3));
          const int k = base + rank;
          // hit predicate re-derived from the ballot (single-use i1s only)
          if (((half >> n) & 1u) && k < K) {
            const int mrow = m0 + v + (halfsel << 3);
            out[(size_t)mrow * K + k] = p;       // to_idx % N
            const size_t cb = (size_t)N * K + ((size_t)mrow * K + k) * 3;
            out[cb + 0] = isx;
            out[cb + 1] = isy;
            out[cb + 2] = isz;
          }
        }
        // advance running counts: row l's ballot half, one bpermute + 2 popc
        const unsigned other = __shfl(myBal, lane & 7);
        const int inc = (lane < 8) ? __popc(other & 0xFFFFu)
                                   : __popc(other >> 16);
        rcount += inc;  // lanes >= 16 hold garbage; never read
      }
      pbase += 16;
      qoff  += 192;
      if (pbase >= N) {
        pbase = 0; ++c; qoff = 12 * n;
        isx = c % 3 - 1; isy = (c / 3) % 3 - 1; isz = c / 9 - 1;
        sxF = L * (float)isx; syF = L * (float)isy; szF = L * (float)isz;
      }
    }
  }
}

extern "C" void kernel_launch(void* const* d_in, const int* in_sizes, int n_in,
                              void* d_out, int out_size, void* d_ws, size_t ws_size,
                              hipStream_t stream) {
  const float* pos   = (const float*)d_in[0];  // [N,3] f32
  const float* cellm = (const float*)d_in[1];  // [3,3] f32 (diagonal L*I)
  const int*   pmaxn = (const int*)d_in[2];    // scalar max_neighbours
  int* out = (int*)d_out;
  const int N = in_sizes[0] / 3;               // 1536

  const int fillBlocks = (out_size + 255) / 256;
  pb_fill<<<fillBlocks, 256, 0, stream>>>(out, pmaxn, N, out_size);
  pb_neigh<<<N / 16, 256, 0, stream>>>(pos, cellm, pmaxn, out, N);
}